// DE_NN_35820027249305
// MI455X (gfx1250) — compile-verified
//
#include <hip/hip_runtime.h>

// MI455X / gfx1250: wave32, WMMA 16x16x4 f32.
// D(16x16) = A(16x4) x B(4x16) + C. Layouts per CDNA5 ISA 7.12.2:
//  A: lane m(0-15) rows M=0..15, VGPR0=K0, VGPR1=K1; lanes 16-31 = K2,K3.
//  B: VGPR0 lanes0-15 = row K0 (N=lane), lanes16-31 = row K2; VGPR1 = K1,K3.
//  C/D: VGPR r, lanes0-15 -> (M=r, N=lane); lanes16-31 -> (M=r+8, N=lane-16).

typedef __attribute__((ext_vector_type(2))) float v2f;
typedef __attribute__((ext_vector_type(8))) float v8f;

__device__ __forceinline__ float bperm(float v, int idxBytes) {
    int r = __builtin_amdgcn_ds_bpermute(idxBytes, __builtin_bit_cast(int, v));
    return __builtin_bit_cast(float, r);
}

__global__ void __launch_bounds__(256)
mlp44_wmma_kernel(const float* __restrict__ X,
                  const float* __restrict__ lin1,
                  const float* __restrict__ lin2,
                  const float* __restrict__ lin3,
                  const float* __restrict__ lin4,
                  float* __restrict__ out,
                  int B)
{
    const int l    = blockIdx.y;                 // which tiny network
    const int lane = threadIdx.x & 31;
    const int wIb  = threadIdx.x >> 5;           // wave in block
    const int m    = lane & 15;
    const bool hi  = lane >= 16;

    const float* w1 = lin1 + l * 4;              // (4,1)
    const float* W2 = lin2 + l * 32;             // (8,4) row-major
    const float* W3 = lin3 + l * 32;             // (4,8) row-major
    const float* w4 = lin4 + l * 4;              // (1,4)

    // ---- hoisted A-matrices (per-l constants) ----
    // A2 = W2 zero-padded to 16x4 (rows 8..15 = 0)
    v2f a2;
    a2.x = (m < 8) ? W2[m * 4 + (hi ? 2 : 0)] : 0.0f;
    a2.y = (m < 8) ? W2[m * 4 + (hi ? 3 : 1)] : 0.0f;
    // A3a = W3[:,0:4], A3b = W3[:,4:8], zero-padded to 16x4 (rows 4..15 = 0)
    v2f a3a, a3b;
    a3a.x = (m < 4) ? W3[m * 8 + (hi ? 2 : 0)] : 0.0f;
    a3a.y = (m < 4) ? W3[m * 8 + (hi ? 3 : 1)] : 0.0f;
    a3b.x = (m < 4) ? W3[m * 8 + 4 + (hi ? 2 : 0)] : 0.0f;
    a3b.y = (m < 4) ? W3[m * 8 + 4 + (hi ? 3 : 1)] : 0.0f;
    // layer-1 weights selected per lane-half (rows {0,1} lo / {2,3} hi)
    const float w1a = w1[hi ? 2 : 0];
    const float w1b = w1[hi ? 3 : 1];
    const float w40 = w4[0], w41 = w4[1], w42 = w4[2], w43 = w4[3];

    const int idxLo = m * 4;                     // bpermute: fetch lane (lane&15)

    const float* xrow = X   + (size_t)l * (size_t)B;
    float*       orow = out + (size_t)l * (size_t)B;

    const int wid        = blockIdx.x * (blockDim.x >> 5) + wIb;
    const int waveStride = gridDim.x * (blockDim.x >> 5);

    // Main loop: 32 columns per iteration (two 16-wide WMMA tiles).
    // Loop bounds are wave-uniform -> EXEC stays all-ones for WMMA.
    for (int base = wid * 32; base + 32 <= B; base += waveStride * 32) {
        const float x0 = xrow[base + m];          // tile 0: cols base..base+15
        const float x1 = xrow[base + 16 + m];     // tile 1: cols base+16..base+31

        float y[2];
        #pragma unroll
        for (int t = 0; t < 2; ++t) {
            const float x = t ? x1 : x0;

            // layer 1 (VALU), built directly in B-matrix layout
            v2f b2;
            b2.x = fmaxf(w1a * x, 0.0f);
            b2.y = fmaxf(w1b * x, 0.0f);

            // layer 2: h2 = relu(W2 @ h1)
            v8f c0 = {};
            v8f d2 = __builtin_amdgcn_wmma_f32_16x16x4_f32(
                false, a2, false, b2, (short)0, c0, false, false);
            #pragma unroll
            for (int r = 0; r < 8; ++r) d2[r] = fmaxf(d2[r], 0.0f);

            // D2 -> B3 layout: lanes16-31 of each B VGPR gather lanes0-15
            // of another D2 VGPR. Execute bpermute with full EXEC, then select.
            const float p2 = bperm(d2[2], idxLo);
            const float p3 = bperm(d2[3], idxLo);
            const float p6 = bperm(d2[6], idxLo);
            const float p7 = bperm(d2[7], idxLo);
            v2f b3a, b3b;
            b3a.x = hi ? p2 : d2[0];
            b3a.y = hi ? p3 : d2[1];
            b3b.x = hi ? p6 : d2[4];
            b3b.y = hi ? p7 : d2[5];

            // layer 3: K=8 as two chained K=4 WMMAs
            v8f d3 = __builtin_amdgcn_wmma_f32_16x16x4_f32(
                false, a3a, false, b3a, (short)0, c0, false, false);
            d3 = __builtin_amdgcn_wmma_f32_16x16x4_f32(
                false, a3b, false, b3b, (short)0, d3, false, false);

            // layer 4 (VALU): rows of h3 are same-lane across d3[0..3]
            const float h0 = fmaxf(d3[0], 0.0f);
            const float h1 = fmaxf(d3[1], 0.0f);
            const float h2 = fmaxf(d3[2], 0.0f);
            const float h3 = fmaxf(d3[3], 0.0f);
            y[t] = w40 * h0 + w41 * h1 + w42 * h2 + w43 * h3; // valid lanes 0-15
        }

        // pack two 16-col tiles into one coalesced 32-lane store
        const float yp = bperm(y[1], idxLo);
        orow[base + lane] = hi ? yp : y[0];
    }

    // Tail (B % 32 != 0): WMMA-free scalar path; empty for B = 400000.
    if (blockIdx.x == 0 && wIb == 0) {
        const int tail = B & ~31;
        for (int col = tail + lane; col < B; col += 32) {
            const float x = xrow[col];
            float h1v[4], h2v[8], h3v[4];
            #pragma unroll
            for (int k = 0; k < 4; ++k) h1v[k] = fmaxf(w1[k] * x, 0.0f);
            #pragma unroll
            for (int j = 0; j < 8; ++j) {
                float s = 0.0f;
                #pragma unroll
                for (int k = 0; k < 4; ++k) s += W2[j * 4 + k] * h1v[k];
                h2v[j] = fmaxf(s, 0.0f);
            }
            #pragma unroll
            for (int j = 0; j < 4; ++j) {
                float s = 0.0f;
                #pragma unroll
                for (int k = 0; k < 8; ++k) s += W3[j * 8 + k] * h2v[k];
                h3v[j] = fmaxf(s, 0.0f);
            }
            float yv = 0.0f;
            #pragma unroll
            for (int k = 0; k < 4; ++k) yv += w4[k] * h3v[k];
            orow[col] = yv;
        }
    }
}

extern "C" void kernel_launch(void* const* d_in, const int* in_sizes, int n_in,
                              void* d_out, int out_size, void* d_ws, size_t ws_size,
                              hipStream_t stream) {
    const float* X    = (const float*)d_in[0];
    const float* lin1 = (const float*)d_in[1];
    const float* lin2 = (const float*)d_in[2];
    const float* lin3 = (const float*)d_in[3];
    const float* lin4 = (const float*)d_in[4];
    float* out = (float*)d_out;

    const int NP = in_sizes[1] / 4;          // lin1s is (NP,4,1)
    const int B  = in_sizes[0] / NP;         // X is (NP,1,B)

    dim3 grid(128, NP);                      // 128 blocks x 8 waves grid-stride per network
    dim3 block(256);
    mlp44_wmma_kernel<<<grid, block, 0, stream>>>(X, lin1, lin2, lin3, lin4, out, B);
}